// RoPE_47107201302970
// MI455X (gfx1250) — compile-verified
//
#include <hip/hip_runtime.h>
#include <math.h>

// Raw ext_vector_type so __builtin_nontemporal_load/store accept it.
typedef float v4f __attribute__((ext_vector_type(4)));

// log2(10000) / 32, so inv_freq_k = 10000^(-k/32) = exp2(-k * C)
#define ROPE_LOG2_THETA_OVER_32 0.4152410118608308f

// Rows of (batch*head) processed per thread in the hot kernel: one table
// entry read serves ROWS_PER_THREAD * 32 streamed bytes.
#define ROWS_PER_THREAD 4

// ---------------------------------------------------------------------------
// Kernel 1 (cold): build cos/sin table in d_ws.
// Entry (s, j), j in [0,16): covers dims 4j..4j+3 = pairs k0=2j, k1=2j+1.
// tbl[s*16 + j] = { cos(a0), sin(a0), cos(a1), sin(a1) }.
// 8192*16 entries * 16 B = 2 MB -> L2 resident, reused by all 128 (b,h) rows.
// ---------------------------------------------------------------------------
__global__ __launch_bounds__(256) void rope_build_table(
    const int* __restrict__ pos, v4f* __restrict__ tbl, int seq)
{
    int idx = blockIdx.x * 256 + threadIdx.x;      // [0, seq*16)
    if (idx >= seq * 16) return;
    int s = idx >> 4;
    int j = idx & 15;

    float p  = (float)pos[s];
    float k0 = (float)(2 * j);
    float f0 = exp2f(-k0 * ROPE_LOG2_THETA_OVER_32);
    float f1 = exp2f(-(k0 + 1.0f) * ROPE_LOG2_THETA_OVER_32);

    float s0, c0, s1, c1;
    sincosf(p * f0, &s0, &c0);                     // accurate range reduction,
    sincosf(p * f1, &s1, &c1);                     // only 131K threads total

    v4f e;
    e.x = c0; e.y = s0; e.z = c1; e.w = s1;
    tbl[idx] = e;
}

// ---------------------------------------------------------------------------
// Kernel 2 (hot): pure HBM stream at NT temporal hint.
// Each thread: 1 table read (RT, L2 hit) + ROWS_PER_THREAD independent NT
// b128 loads/stores at stride seq16 (the batch*head axis). The table entry
// is rotated into registers once and reused, and the 4 loads are in flight
// simultaneously for MLP.
// grid = (seq16/256, rows/ROWS_PER_THREAD).
// ---------------------------------------------------------------------------
__global__ __launch_bounds__(256) void rope_apply_tbl(
    const v4f* __restrict__ x, const v4f* __restrict__ tbl,
    v4f* __restrict__ out, int seq16, int rows)
{
    int l = blockIdx.x * 256 + threadIdx.x;        // [0, seq16)
    if (l >= seq16) return;

    v4f e = tbl[l];                                // RT -> L2/WGP$ resident

    int r0 = blockIdx.y * ROWS_PER_THREAD;
    size_t stride = (size_t)seq16;
    size_t g = (size_t)r0 * stride + (size_t)l;

    int nr = rows - r0;
    if (nr > ROWS_PER_THREAD) nr = ROWS_PER_THREAD;

    if (nr == ROWS_PER_THREAD) {
        // Fast path: fully unrolled, 4 NT loads issued before any use.
        v4f v0 = __builtin_nontemporal_load(x + g);
        v4f v1 = __builtin_nontemporal_load(x + g + stride);
        v4f v2 = __builtin_nontemporal_load(x + g + 2 * stride);
        v4f v3 = __builtin_nontemporal_load(x + g + 3 * stride);

        v4f o0, o1, o2, o3;
        o0.x = fmaf(e.x, v0.x, -(e.y * v0.y));
        o0.y = fmaf(e.y, v0.x,  (e.x * v0.y));
        o0.z = fmaf(e.z, v0.z, -(e.w * v0.w));
        o0.w = fmaf(e.w, v0.z,  (e.z * v0.w));

        o1.x = fmaf(e.x, v1.x, -(e.y * v1.y));
        o1.y = fmaf(e.y, v1.x,  (e.x * v1.y));
        o1.z = fmaf(e.z, v1.z, -(e.w * v1.w));
        o1.w = fmaf(e.w, v1.z,  (e.z * v1.w));

        o2.x = fmaf(e.x, v2.x, -(e.y * v2.y));
        o2.y = fmaf(e.y, v2.x,  (e.x * v2.y));
        o2.z = fmaf(e.z, v2.z, -(e.w * v2.w));
        o2.w = fmaf(e.w, v2.z,  (e.z * v2.w));

        o3.x = fmaf(e.x, v3.x, -(e.y * v3.y));
        o3.y = fmaf(e.y, v3.x,  (e.x * v3.y));
        o3.z = fmaf(e.z, v3.z, -(e.w * v3.w));
        o3.w = fmaf(e.w, v3.z,  (e.z * v3.w));

        __builtin_nontemporal_store(o0, out + g);
        __builtin_nontemporal_store(o1, out + g + stride);
        __builtin_nontemporal_store(o2, out + g + 2 * stride);
        __builtin_nontemporal_store(o3, out + g + 3 * stride);
    } else {
        for (int r = 0; r < nr; ++r) {
            v4f v = __builtin_nontemporal_load(x + g);
            v4f o;
            o.x = fmaf(e.x, v.x, -(e.y * v.y));
            o.y = fmaf(e.y, v.x,  (e.x * v.y));
            o.z = fmaf(e.z, v.z, -(e.w * v.w));
            o.w = fmaf(e.w, v.z,  (e.z * v.w));
            __builtin_nontemporal_store(o, out + g);
            g += stride;
        }
    }
}

// ---------------------------------------------------------------------------
// Fallback (only if ws_size < 2 MB): trig inline, reused across 4 rows.
// ---------------------------------------------------------------------------
__global__ __launch_bounds__(256) void rope_apply_inline(
    const v4f* __restrict__ x, const int* __restrict__ pos,
    v4f* __restrict__ out, int seq16, int rows)
{
    int l = blockIdx.x * 256 + threadIdx.x;
    if (l >= seq16) return;

    int s = l >> 4;
    int j = l & 15;
    float p  = (float)pos[s];
    float k0 = (float)(2 * j);
    float f0 = exp2f(-k0 * ROPE_LOG2_THETA_OVER_32);
    float f1 = exp2f(-(k0 + 1.0f) * ROPE_LOG2_THETA_OVER_32);
    float s0, c0, s1, c1;
    sincosf(p * f0, &s0, &c0);
    sincosf(p * f1, &s1, &c1);

    int r0 = blockIdx.y * ROWS_PER_THREAD;
    size_t stride = (size_t)seq16;
    size_t g = (size_t)r0 * stride + (size_t)l;
    int nr = rows - r0;
    if (nr > ROWS_PER_THREAD) nr = ROWS_PER_THREAD;

    for (int r = 0; r < nr; ++r) {
        v4f v = __builtin_nontemporal_load(x + g);
        v4f o;
        o.x = c0 * v.x - s0 * v.y;
        o.y = s0 * v.x + c0 * v.y;
        o.z = c1 * v.z - s1 * v.w;
        o.w = s1 * v.z + c1 * v.w;
        __builtin_nontemporal_store(o, out + g);
        g += stride;
    }
}

extern "C" void kernel_launch(void* const* d_in, const int* in_sizes, int n_in,
                              void* d_out, int out_size, void* d_ws, size_t ws_size,
                              hipStream_t stream) {
    const float* x   = (const float*)d_in[0];
    const int*   pos = (const int*)d_in[1];
    float*       out = (float*)d_out;

    const int d_k   = 64;                      // reference D_K
    const int seq   = in_sizes[1];             // 8192 token positions
    const int seq16 = seq * (d_k / 4);         // float4s per (batch,head) row
    const long long n4 = (long long)out_size / 4;
    const int  rows = (int)(n4 / (long long)seq16);   // batch*heads = 128

    const size_t tbl_bytes = (size_t)seq16 * 4 * sizeof(float);  // 2 MB

    dim3 block(256);
    dim3 grid((seq16 + 255) / 256,
              (rows + ROWS_PER_THREAD - 1) / ROWS_PER_THREAD);

    if (ws_size >= tbl_bytes) {
        // Build cos/sin table (tiny), then stream. Same stream -> in-order,
        // no sync needed; both safe for graph capture.
        rope_build_table<<<dim3((seq * 16 + 255) / 256), block, 0, stream>>>(
            pos, (v4f*)d_ws, seq);
        rope_apply_tbl<<<grid, block, 0, stream>>>(
            (const v4f*)x, (const v4f*)d_ws, (v4f*)out, seq16, rows);
    } else {
        rope_apply_inline<<<grid, block, 0, stream>>>(
            (const v4f*)x, pos, (v4f*)out, seq16, rows);
    }
}